// Rasterer_67104569033240
// MI455X (gfx1250) — compile-verified
//
#include <hip/hip_runtime.h>
#include <hip/hip_bf16.h>
#include <math.h>

// ---------------------------------------------------------------------------
// Rasterizer on MI455X (gfx1250): edge functions are affine in pixel coords,
// so (16 edges) x (16 pixels) evaluation = (16x4)@(4x16) fp32 GEMM
// -> V_WMMA_F32_16X16X4_F32.  Triangles padded to 4 edge rows each, so one
// WMMA block = 4 triangles x 16 pixels; each lane yields 2 triangle scores.
// ---------------------------------------------------------------------------

#define RES 256
#define N_PIX (RES * RES)          // 65536
#define N_MODELS 8
#define N_TRI 1000
#define BATCH 2
#define EPSF 1e-8f

#define ROWS_PER_TRI 4             // 3 real edges + 1 pad row
#define ROWS_PER_BATCH (N_TRI * ROWS_PER_TRI)   // 4000
#define EDGE_BLOCKS (ROWS_PER_BATCH / 16)       // 250 (4 tris per block)
#define PIX_BLOCKS (N_PIX / 16)                 // 4096
#define SLICES 16
#define PB_PER_SLICE (PIX_BLOCKS / SLICES)      // 256

typedef float v2f __attribute__((ext_vector_type(2)));
typedef float v8f __attribute__((ext_vector_type(8)));

__device__ __forceinline__ float relu_(float v) { return fmaxf(v, 0.0f); }

// --------------------------------------------------------------------------
// Kernel 0: init per-triangle min/max accumulators (scores are >= 0, so we
// can use unsigned-bit ordering: min init = +inf bits, max init = 0).
// --------------------------------------------------------------------------
__global__ void init_minmax_kernel(unsigned* mnBits, unsigned* mxBits) {
    int tid = blockIdx.x * blockDim.x + threadIdx.x;
    if (tid < BATCH * N_TRI) {
        mnBits[tid] = 0x7F800000u;   // +inf
        mxBits[tid] = 0u;            // +0.0
    }
}

// --------------------------------------------------------------------------
// Kernel 1: per (batch, triangle) compute projected vertices and edge-function
// coefficients t_i(px,py) = cpx*px + cpy*py + cc  (already scaled by N).
// Row layout (float4 per row): tri t -> rows 4t..4t+3, row 3 = zero pad.
// --------------------------------------------------------------------------
__global__ void setup_coeffs_kernel(const float* __restrict__ meshes,
                                    const float* __restrict__ K,
                                    const int*   __restrict__ model_idxs,
                                    const float* __restrict__ cams,
                                    float* __restrict__ coef) {
    int tid = blockIdx.x * blockDim.x + threadIdx.x;
    if (tid >= BATCH * N_TRI) return;
    int b = tid / N_TRI;
    int t = tid % N_TRI;

    const float* cam = cams + b * 12;          // 3x4
    // KC = K @ cam  (3x4)
    float KC[12];
    #pragma unroll
    for (int i = 0; i < 3; ++i)
        #pragma unroll
        for (int j = 0; j < 4; ++j) {
            float s = 0.0f;
            #pragma unroll
            for (int k = 0; k < 3; ++k) s += K[i * 3 + k] * cam[k * 4 + j];
            KC[i * 4 + j] = s;
        }

    int m = model_idxs[b];
    const float* v = meshes + ((size_t)m * N_TRI + t) * 9;

    float x[3], y[3];
    #pragma unroll
    for (int i = 0; i < 3; ++i) {
        float X = v[i * 3 + 0], Y = v[i * 3 + 1], Z = v[i * 3 + 2];
        float u  = KC[0] * X + KC[1] * Y + KC[2]  * Z + KC[3];
        float vv = KC[4] * X + KC[5] * Y + KC[6]  * Z + KC[7];
        float w  = KC[8] * X + KC[9] * Y + KC[10] * Z + KC[11];
        float inv = 1.0f / (w + EPSF);
        x[i] = u * inv;
        y[i] = vv * inv;
    }

    float ex[3], ey[3];
    ex[0] = x[1] - x[0]; ey[0] = y[1] - y[0];   // e0 paired with vertex 0
    ex[1] = x[2] - x[1]; ey[1] = y[2] - y[1];   // e1 paired with vertex 1
    ex[2] = x[0] - x[2]; ey[2] = y[0] - y[2];   // e2 paired with vertex 2
    float Nv = ex[0] * ey[2] - ey[0] * ex[2] + EPSF;

    float* o = coef + (((size_t)b * ROWS_PER_BATCH) + (size_t)t * ROWS_PER_TRI) * 4;
    #pragma unroll
    for (int i = 0; i < 3; ++i) {
        // N*(e.x*(v.y-py) - e.y*(v.x-px)) = (N*e.y)*px + (-N*e.x)*py + N*(e.x*v.y - e.y*v.x)
        o[i * 4 + 0] = Nv * ey[i];
        o[i * 4 + 1] = -Nv * ex[i];
        o[i * 4 + 2] = Nv * (ex[i] * y[i] - ey[i] * x[i]);
        o[i * 4 + 3] = 0.0f;
    }
    o[12] = 0.0f; o[13] = 0.0f; o[14] = 0.0f; o[15] = 0.0f;  // pad row
}

// --------------------------------------------------------------------------
// WMMA helper: evaluate 16 edge rows (A frag, loaded by caller) at the 16
// pixels of pixel-block pb.  Basis rows: 0=px, 1=py, 2=1, 3=0.
// B 4x16 f32 layout: VGPR0 = row0 (lanes 0-15) / row2 (lanes 16-31),
//                    VGPR1 = row1 / row3.
// --------------------------------------------------------------------------
__device__ __forceinline__ v8f eval_block(v2f afrag, int pb, int n, bool low) {
    int P = pb * 16 + n;
    float px = (float)(P >> 8);
    float py = (float)(P & 255);
    v2f bfrag;
    bfrag.x = low ? px : 1.0f;
    bfrag.y = low ? py : 0.0f;
    v8f c = {0.f, 0.f, 0.f, 0.f, 0.f, 0.f, 0.f, 0.f};
    return __builtin_amdgcn_wmma_f32_16x16x4_f32(
        false, afrag, false, bfrag, (short)0, c, false, false);
}

// --------------------------------------------------------------------------
// Kernel 2 (pass A): per-triangle min/max of score over all pixels.
// One wave = (batch, edge-block, pixel-slice); 256 WMMAs per wave.
// --------------------------------------------------------------------------
__global__ void minmax_kernel(const float* __restrict__ coef,
                              unsigned* __restrict__ mnBits,
                              unsigned* __restrict__ mxBits) {
    int wave = (blockIdx.x * blockDim.x + threadIdx.x) >> 5;
    int lane = threadIdx.x & 31;
    int b     = wave / (EDGE_BLOCKS * SLICES);
    int rem   = wave % (EDGE_BLOCKS * SLICES);
    int eb    = rem / SLICES;
    int slice = rem % SLICES;

    int n = lane & 15;
    bool low = lane < 16;

    // A fragment (16x4 f32): lane holds row M = lane%16; VGPR0/1 = K {0,1}
    // for lanes 0-15 and K {2,3} for lanes 16-31.
    int row = eb * 16 + n;
    const float* cp = coef + (((size_t)b * ROWS_PER_BATCH) + row) * 4 + (low ? 0 : 2);
    v2f afrag; afrag.x = cp[0]; afrag.y = cp[1];

    float mn0 = __uint_as_float(0x7F800000u), mx0 = 0.0f;
    float mn1 = __uint_as_float(0x7F800000u), mx1 = 0.0f;

    for (int i = 0; i < PB_PER_SLICE; ++i) {
        int pb = slice * PB_PER_SLICE + i;
        v8f d = eval_block(afrag, pb, n, low);
        float s0 = relu_(d[0]) * relu_(d[1]) * relu_(d[2]);  // tri pair, rows 0-2
        float s1 = relu_(d[4]) * relu_(d[5]) * relu_(d[6]);  // rows 4-6
        mn0 = fminf(mn0, s0); mx0 = fmaxf(mx0, s0);
        mn1 = fminf(mn1, s1); mx1 = fmaxf(mx1, s1);
    }

    // reduce over the 16 lanes of each half (same triangles, different pixels)
    #pragma unroll
    for (int m = 8; m >= 1; m >>= 1) {
        mn0 = fminf(mn0, __shfl_xor(mn0, m, 32));
        mx0 = fmaxf(mx0, __shfl_xor(mx0, m, 32));
        mn1 = fminf(mn1, __shfl_xor(mn1, m, 32));
        mx1 = fmaxf(mx1, __shfl_xor(mx1, m, 32));
    }
    if (n == 0) {
        // low half -> tris eb*4 + {0,1}; high half -> tris eb*4 + {2,3}
        int tri = b * N_TRI + eb * 4 + (low ? 0 : 2);
        atomicMin(&mnBits[tri],     __float_as_uint(mn0));
        atomicMax(&mxBits[tri],     __float_as_uint(mx0));
        atomicMin(&mnBits[tri + 1], __float_as_uint(mn1));
        atomicMax(&mxBits[tri + 1], __float_as_uint(mx1));
    }
}

// --------------------------------------------------------------------------
// Kernel 3 (pass B): per-pixel sum of tanh-normalized scores over all tris.
// One wave = (batch, pixel-block); loops over 250 edge-blocks (250 WMMAs).
// --------------------------------------------------------------------------
__global__ void render_kernel(const float* __restrict__ coef,
                              const unsigned* __restrict__ mnBits,
                              const unsigned* __restrict__ mxBits,
                              float* __restrict__ out) {
    int wave = (blockIdx.x * blockDim.x + threadIdx.x) >> 5;
    int lane = threadIdx.x & 31;
    int b  = wave / PIX_BLOCKS;
    int pb = wave % PIX_BLOCKS;

    int n = lane & 15;
    bool low = lane < 16;

    float acc = 0.0f;
    for (int eb = 0; eb < EDGE_BLOCKS; ++eb) {
        int row = eb * 16 + n;
        const float* cp = coef + (((size_t)b * ROWS_PER_BATCH) + row) * 4 + (low ? 0 : 2);
        v2f afrag; afrag.x = cp[0]; afrag.y = cp[1];

        v8f d = eval_block(afrag, pb, n, low);
        float s0 = relu_(d[0]) * relu_(d[1]) * relu_(d[2]);
        float s1 = relu_(d[4]) * relu_(d[5]) * relu_(d[6]);

        int tri = b * N_TRI + eb * 4 + (low ? 0 : 2);
        float mnA = __uint_as_float(mnBits[tri]);
        float mxA = __uint_as_float(mxBits[tri]);
        float mnB = __uint_as_float(mnBits[tri + 1]);
        float mxB = __uint_as_float(mxBits[tri + 1]);

        acc += tanhf((s0 - mnA) / (mxA - mnA + EPSF));
        acc += tanhf((s1 - mnB) / (mxB - mnB + EPSF));
    }
    // lane L (tris 0/1) + lane L+16 (tris 2/3) cover the same pixel
    acc += __shfl_xor(acc, 16, 32);
    if (low) {
        int P = pb * 16 + n;
        out[(size_t)b * N_PIX + P] = acc;
    }
}

// --------------------------------------------------------------------------
// Launch: inputs = {meshes(f32), K(f32), model_idxs(i32), camera_poses(f32)}.
// Workspace: [coef: BATCH*4000*4 f32 = 128000 B][mn: 2000 u32][mx: 2000 u32].
// --------------------------------------------------------------------------
extern "C" void kernel_launch(void* const* d_in, const int* in_sizes, int n_in,
                              void* d_out, int out_size, void* d_ws, size_t ws_size,
                              hipStream_t stream) {
    const float* meshes = (const float*)d_in[0];
    const float* K      = (const float*)d_in[1];
    const int*   idxs   = (const int*)d_in[2];
    const float* cams   = (const float*)d_in[3];
    float* out = (float*)d_out;

    float*    coef   = (float*)d_ws;
    unsigned* mnBits = (unsigned*)((char*)d_ws + 131072);
    unsigned* mxBits = (unsigned*)((char*)d_ws + 131072 + 8192);

    // 0) init min/max accumulators
    init_minmax_kernel<<<(BATCH * N_TRI + 255) / 256, 256, 0, stream>>>(mnBits, mxBits);
    // 1) edge-function coefficients
    setup_coeffs_kernel<<<(BATCH * N_TRI + 255) / 256, 256, 0, stream>>>(
        meshes, K, idxs, cams, coef);
    // 2) per-triangle min/max over pixels (8000 waves = 1000 blocks x 8 waves)
    minmax_kernel<<<(BATCH * EDGE_BLOCKS * SLICES) / 8, 256, 0, stream>>>(
        coef, mnBits, mxBits);
    // 3) tanh-normalized per-pixel sums (8192 waves = 1024 blocks x 8 waves)
    render_kernel<<<(BATCH * PIX_BLOCKS) / 8, 256, 0, stream>>>(
        coef, mnBits, mxBits, out);
}